// ReEig_34583076667573
// MI455X (gfx1250) — compile-verified
//
#include <hip/hip_runtime.h>

// ReEig via matrix-sign Newton-Schulz, fully on V_WMMA_F32_16X16X4_F32.
//   Y = X - eps*I ;  rec = eps*I + 0.5*(Y + sign(Y)*Y)
//   sign(Y): Z0 = Y/||Y||_F ;  Z <- Z*(1.5*I - 0.5*Z*Z)   (20 iters)
//
// LDS "layout L": element (i,j) stored at (i>>1)*PS + 2*j + (i&1).
//  - B operand rows (k,k+1) of one column -> one ds_load_b64
//  - A operand (via symmetry A[m][k]=P[k][m]) -> one ds_load_b64
//  - PS = 160 dwords (== 32 mod 64): the half-wave offset h*PS lands lanes
//    16-31 on banks 32..63 -> operand loads hit all 64 LDS banks exactly once.

#define EPS      1e-4f
#define NS_ITERS 20
#define PS       160            // dwords per interleaved row-pair
#define MSZ      (32 * PS)      // 5120 floats = 20 KB per matrix buffer

typedef float v2f __attribute__((ext_vector_type(2)));
typedef float v8f __attribute__((ext_vector_type(8)));

// acc += Pa[m0:m0+16, 0:64] * Pb[0:64, n0:n0+16]   (16 x V_WMMA_F32_16X16X4_F32)
// Pa must be symmetric (A is read transposed through layout L); Pb read naturally.
__device__ __forceinline__ void mm64(const float* __restrict__ Pa,
                                     const float* __restrict__ Pb,
                                     int m0, int n0, int lane, v8f& acc)
{
    const int h    = lane >> 4;          // half-wave: K sub-pair select
    const int c16  = lane & 15;
    const int aoff = h * PS + 2 * (m0 + c16);   // A[m][k0+2h+v] = Pa[k0+2h+v][m]
    const int boff = h * PS + 2 * (n0 + c16);   // B[k0+2h+v][n]
#pragma unroll
    for (int k0 = 0; k0 < 64; k0 += 4) {
        const int kb = (k0 >> 1) * PS;
        const v2f a = *(const v2f*)(Pa + kb + aoff);
        const v2f b = *(const v2f*)(Pb + kb + boff);
        acc = __builtin_amdgcn_wmma_f32_16x16x4_f32(false, a, false, b,
                                                    (short)0, acc, false, false);
    }
}

__global__ __launch_bounds__(256)
void reeig_ns_kernel(const float* __restrict__ X, float* __restrict__ Out, int nmat)
{
    __shared__ float Ybuf[MSZ];   // Y = X - eps*I (kept for final step)
    __shared__ float Zbuf[MSZ];   // NS iterate (symmetric)
    __shared__ float Tbuf[MSZ];   // 1.5*I - 0.5*Z*Z
    __shared__ float snorm;

    const int b    = blockIdx.x;
    const int tid  = threadIdx.x;
    const int lane = tid & 31;
    const int wave = tid >> 5;

    const float* __restrict__ Xb = X + (size_t)b * 4096;

    // ---- load X as row-pairs, form Y = X - eps*I in layout L, accumulate ||Y||_F^2 ----
    if (tid == 0) snorm = 0.f;
    const int p  = tid >> 3;            // row pair 0..31  (rows 2p, 2p+1)
    const int cb = (tid & 7) << 3;      // column block of 8
    float ss = 0.f;
    float4 a0 = *(const float4*)(Xb + (2 * p) * 64 + cb);
    float4 a1 = *(const float4*)(Xb + (2 * p) * 64 + cb + 4);
    float4 b0 = *(const float4*)(Xb + (2 * p + 1) * 64 + cb);
    float4 b1 = *(const float4*)(Xb + (2 * p + 1) * 64 + cb + 4);
    float r0[8] = {a0.x, a0.y, a0.z, a0.w, a1.x, a1.y, a1.z, a1.w};
    float r1[8] = {b0.x, b0.y, b0.z, b0.w, b1.x, b1.y, b1.z, b1.w};
#pragma unroll
    for (int j = 0; j < 8; ++j) {
        const int c = cb + j;
        const float e0 = r0[j] - (2 * p     == c ? EPS : 0.f);
        const float e1 = r1[j] - (2 * p + 1 == c ? EPS : 0.f);
        ss += e0 * e0 + e1 * e1;
        float2 e; e.x = e0; e.y = e1;
        *(float2*)(Ybuf + p * PS + 2 * c) = e;     // ds_store_b64
    }
#pragma unroll
    for (int off = 16; off > 0; off >>= 1) ss += __shfl_down(ss, off, 32);
    __syncthreads();                      // snorm init visible
    if (lane == 0) atomicAdd(&snorm, ss); // ds_add_f32
    __syncthreads();

    const float inv = __frsqrt_rn(snorm + 1e-30f);   // 1/||Y||_F
#pragma unroll
    for (int j = 0; j < 8; ++j) {
        const int o = p * PS + 2 * (cb + j);
        float2 y = *(const float2*)(Ybuf + o);
        y.x *= inv; y.y *= inv;
        *(float2*)(Zbuf + o) = y;
    }
    __syncthreads();

    // tile ownership: wave w -> row band m0, two column tiles n0, n0+16
    const int m0  = (wave >> 1) * 16;
    const int n0  = (wave & 1) * 32;
    const int h   = lane >> 4;
    const int c16 = lane & 15;
    const int nc  = n0 + c16;

    // ---- Newton-Schulz: Z <- Z * (1.5 I - 0.5 Z^2) ----
    for (int it = 0; it < NS_ITERS; ++it) {
        v8f t0 = {}, t1 = {};
        mm64(Zbuf, Zbuf, m0, n0,      lane, t0);
        mm64(Zbuf, Zbuf, m0, n0 + 16, lane, t1);
#pragma unroll
        for (int r = 0; r < 8; r += 2) {            // accumulator rows r,r+1 are adjacent in L
            const int me = m0 + 8 * h + r;
            const int o  = (me >> 1) * PS;
            float2 e0, e1;
            e0.x = (me     == nc      ? 1.5f : 0.f) - 0.5f * t0[r];
            e0.y = (me + 1 == nc      ? 1.5f : 0.f) - 0.5f * t0[r + 1];
            e1.x = (me     == nc + 16 ? 1.5f : 0.f) - 0.5f * t1[r];
            e1.y = (me + 1 == nc + 16 ? 1.5f : 0.f) - 0.5f * t1[r + 1];
            *(float2*)(Tbuf + o + 2 * nc)        = e0;
            *(float2*)(Tbuf + o + 2 * (nc + 16)) = e1;
        }
        __syncthreads();                  // T complete

        v8f w0 = {}, w1 = {};
        mm64(Zbuf, Tbuf, m0, n0,      lane, w0);
        mm64(Zbuf, Tbuf, m0, n0 + 16, lane, w1);
        __syncthreads();                  // all reads of Zbuf done (band shared by wave pairs)
#pragma unroll
        for (int r = 0; r < 8; r += 2) {
            const int me = m0 + 8 * h + r;
            const int o  = (me >> 1) * PS;
            float2 e0, e1;
            e0.x = w0[r];     e0.y = w0[r + 1];
            e1.x = w1[r];     e1.y = w1[r + 1];
            *(float2*)(Zbuf + o + 2 * nc)        = e0;
            *(float2*)(Zbuf + o + 2 * (nc + 16)) = e1;
        }
        __syncthreads();                  // Z updated
    }

    // ---- rec = eps*I + 0.5*(Y + sign(Y)*Y) ----
    v8f q0 = {}, q1 = {};
    mm64(Zbuf, Ybuf, m0, n0,      lane, q0);
    mm64(Zbuf, Ybuf, m0, n0 + 16, lane, q1);

    float* __restrict__ Ob = Out + (size_t)b * 4096;
#pragma unroll
    for (int r = 0; r < 8; r += 2) {
        const int me = m0 + 8 * h + r;
        const int o  = (me >> 1) * PS;
        const float2 y0 = *(const float2*)(Ybuf + o + 2 * nc);         // Y[me][nc], Y[me+1][nc]
        const float2 y1 = *(const float2*)(Ybuf + o + 2 * (nc + 16));
        Ob[(me    ) * 64 + nc]      = 0.5f * (y0.x + q0[r])     + (me     == nc      ? EPS : 0.f);
        Ob[(me + 1) * 64 + nc]      = 0.5f * (y0.y + q0[r + 1]) + (me + 1 == nc      ? EPS : 0.f);
        Ob[(me    ) * 64 + nc + 16] = 0.5f * (y1.x + q1[r])     + (me     == nc + 16 ? EPS : 0.f);
        Ob[(me + 1) * 64 + nc + 16] = 0.5f * (y1.y + q1[r + 1]) + (me + 1 == nc + 16 ? EPS : 0.f);
    }
    (void)nmat;
}

extern "C" void kernel_launch(void* const* d_in, const int* in_sizes, int n_in,
                              void* d_out, int out_size, void* d_ws, size_t ws_size,
                              hipStream_t stream)
{
    const float* x   = (const float*)d_in[0];
    float*       out = (float*)d_out;
    const int nmat = in_sizes[0] / (64 * 64);   // 8192
    reeig_ns_kernel<<<nmat, 256, 0, stream>>>(x, out, nmat);
    (void)n_in; (void)out_size; (void)d_ws; (void)ws_size;
}